// TtMoeLayer_55104430408092
// MI455X (gfx1250) — compile-verified
//
#include <hip/hip_runtime.h>
#include <math.h>

// Problem dims (match reference)
#define B_TOK 32
#define HDIM  4096
#define FDIM  4096
#define EDIM  8
#define TOPK  2

typedef __attribute__((ext_vector_type(2))) float v2f;
typedef __attribute__((ext_vector_type(8))) float v8f;

// D = A(16x4) * B(4x16) + C(16x16), fp32 WMMA (CDNA5)
__device__ __forceinline__ v8f wmma4(v2f a, v2f b, v8f c) {
    return __builtin_amdgcn_wmma_f32_16x16x4_f32(
        /*neg_a=*/false, a, /*neg_b=*/false, b,
        /*c_mod=*/(short)0, c, /*reuse_a=*/false, /*reuse_b=*/false);
}

// ---------------------------------------------------------------------------
// Kernel 1: logits[b][e] = sum_h x[b][h] * Wg[h][e]    (32 blocks x 256 thr)
// ---------------------------------------------------------------------------
__global__ void gate_logits_kernel(const float* __restrict__ x,
                                   const float* __restrict__ Wg,
                                   float* __restrict__ logits) {
    const int b = blockIdx.x;
    const int t = threadIdx.x;
    const int e = t & 7;          // expert
    const int seg = t >> 3;       // 0..31, each covers 128 h's
    const float* xb = x + (size_t)b * HDIM + seg * 128;
    const float* wg = Wg + (size_t)seg * 128 * EDIM + e;
    float acc = 0.f;
#pragma unroll 8
    for (int j = 0; j < 128; ++j) acc += xb[j] * wg[(size_t)j * EDIM];
    __shared__ float part[256];
    part[t] = acc;
    __syncthreads();
    if (t < EDIM) {
        float s = 0.f;
        for (int g = 0; g < 32; ++g) s += part[g * 8 + t];   // fixed order
        logits[b * EDIM + t] = s;
    }
}

// ---------------------------------------------------------------------------
// Kernel 2: top-2 + softmax per token; build per-expert token lists & slots.
// Single block; serial slot assignment in thread 0 for determinism.
// ---------------------------------------------------------------------------
__global__ void gate_route_kernel(const float* __restrict__ logits,
                                  int* __restrict__ cnt,
                                  int* __restrict__ list,
                                  int* __restrict__ tok_e,
                                  int* __restrict__ tok_slot,
                                  float* __restrict__ tok_w) {
    __shared__ int te[B_TOK][2];
    const int t = threadIdx.x;
    if (t < B_TOK) {
        float l[EDIM];
        for (int e = 0; e < EDIM; ++e) l[e] = logits[t * EDIM + e];
        int i0 = 0; float v0 = l[0];
        for (int e = 1; e < EDIM; ++e) if (l[e] > v0) { v0 = l[e]; i0 = e; } // ties -> lowest idx
        int i1 = -1; float v1 = -3.402823466e38f;
        for (int e = 0; e < EDIM; ++e) if (e != i0 && l[e] > v1) { v1 = l[e]; i1 = e; }
        // softmax over {v0, v1}, v0 >= v1
        float ex = expf(v1 - v0);
        float w0 = 1.f / (1.f + ex);
        float w1 = ex * w0;
        tok_e[t * 2 + 0] = i0; tok_e[t * 2 + 1] = i1;
        tok_w[t * 2 + 0] = w0; tok_w[t * 2 + 1] = w1;
        te[t][0] = i0; te[t][1] = i1;
    }
    __syncthreads();
    if (t == 0) {
        int c[EDIM];
        for (int e = 0; e < EDIM; ++e) c[e] = 0;
        for (int b = 0; b < B_TOK; ++b)
            for (int k = 0; k < TOPK; ++k) {
                int e = te[b][k];
                int s = c[e]++;
                list[e * B_TOK + s] = b;
                tok_slot[b * 2 + k] = s;
            }
        for (int e = 0; e < EDIM; ++e) cnt[e] = c[e];
    }
}

// ---------------------------------------------------------------------------
// Kernel 3: per expert, gathered rows:  g = silu(x*W1) * (x*W3)
// Block tile: M=16 (one M-tile of routed tokens), N=128, K-chunk=32.
// 8 waves, each wave owns one 16x16 N-subtile (two accumulators: W1 & W3).
// ---------------------------------------------------------------------------
__global__ void __launch_bounds__(256)
expert_gemm1_kernel(const float* __restrict__ x,
                    const float* __restrict__ W1,
                    const float* __restrict__ W3,
                    const int* __restrict__ cnt,
                    const int* __restrict__ list,
                    float* __restrict__ g_ws) {
    const int e      = blockIdx.z;
    const int tile_m = blockIdx.y;
    const int f0     = blockIdx.x * 128;
    const int nc     = cnt[e];
    if (tile_m * 16 >= nc) return;    // uniform exit: EXEC stays all-ones below

    __shared__ float Alds[16][34];    // pad 34: conflict-free, 8B-aligned pairs
    __shared__ float B1lds[32][128];
    __shared__ float B3lds[32][128];
    __shared__ int   tokrow[16];

    const int t = threadIdx.x;
    if (t < 16) {
        int s = tile_m * 16 + t;
        tokrow[t] = (s < nc) ? list[e * B_TOK + s] : -1;   // -1 => zero pad row
    }
    __syncthreads();

    const int lane = t & 31;
    const int wv   = t >> 5;          // wave 0..7
    const int mi   = lane & 15;
    const int half = lane >> 4;       // 0: K0/K1, 1: K2/K3
    const int n0   = wv * 16;

    // A staging role: row ar (0..15), cols ac, ac+1 of the K-chunk
    const int ar = t >> 4;
    const int ac = (t & 15) * 2;
    const int atok = tokrow[ar];
    const float* xrow = (atok >= 0) ? (x + (size_t)atok * HDIM) : nullptr;

    const size_t wbase = (size_t)e * HDIM * FDIM + f0;

    v8f acc1 = {}; v8f acc3 = {};

    for (int k0 = 0; k0 < HDIM; k0 += 32) {
        float a0 = 0.f, a1 = 0.f;
        if (xrow) { a0 = xrow[k0 + ac]; a1 = xrow[k0 + ac + 1]; }
        float4 bv1[4], bv3[4];
#pragma unroll
        for (int i = 0; i < 4; ++i) {
            int q = t + i * 256;
            int row = q >> 5;             // 0..31
            int col = (q & 31) * 4;       // 0..124
            bv1[i] = *(const float4*)(W1 + wbase + (size_t)(k0 + row) * FDIM + col);
            bv3[i] = *(const float4*)(W3 + wbase + (size_t)(k0 + row) * FDIM + col);
        }
        __syncthreads();                  // previous chunk's compute done
        Alds[ar][ac] = a0; Alds[ar][ac + 1] = a1;
#pragma unroll
        for (int i = 0; i < 4; ++i) {
            int q = t + i * 256;
            int row = q >> 5;
            int col = (q & 31) * 4;
            *(float4*)&B1lds[row][col] = bv1[i];
            *(float4*)&B3lds[row][col] = bv3[i];
        }
        __syncthreads();                  // chunk visible
#pragma unroll
        for (int kk = 0; kk < 8; ++kk) {
            const int kb = kk * 4 + half * 2;     // even => 8B aligned
            v2f a = *(const v2f*)&Alds[mi][kb];
            v2f b1; b1.x = B1lds[kb][n0 + mi]; b1.y = B1lds[kb + 1][n0 + mi];
            v2f b3; b3.x = B3lds[kb][n0 + mi]; b3.y = B3lds[kb + 1][n0 + mi];
            acc1 = wmma4(a, b1, acc1);
            acc3 = wmma4(a, b3, acc3);
        }
    }

    // C/D layout: vgpr r, lanes0-15 -> M=r, lanes16-31 -> M=r+8; N = lane&15
    const size_t gbase = ((size_t)e * B_TOK + tile_m * 16) * FDIM + f0 + n0;
#pragma unroll
    for (int r = 0; r < 8; ++r) {
        const int m = r + half * 8;
        float h1 = acc1[r];
        float h3 = acc3[r];
        float gv = (h1 / (1.f + expf(-h1))) * h3;   // silu(h1) * h3
        g_ws[gbase + (size_t)m * FDIM + mi] = gv;   // padded rows write 0
    }
}

// ---------------------------------------------------------------------------
// Kernel 4: per expert:  y = g * W2   (same tiling, single accumulator)
// ---------------------------------------------------------------------------
__global__ void __launch_bounds__(256)
expert_gemm2_kernel(const float* __restrict__ W2,
                    const int* __restrict__ cnt,
                    const float* __restrict__ g_ws,
                    float* __restrict__ y_ws) {
    const int e      = blockIdx.z;
    const int tile_m = blockIdx.y;
    const int h0     = blockIdx.x * 128;
    const int nc     = cnt[e];
    if (tile_m * 16 >= nc) return;

    __shared__ float Alds[16][34];
    __shared__ float Blds[32][128];

    const int t    = threadIdx.x;
    const int lane = t & 31;
    const int wv   = t >> 5;
    const int mi   = lane & 15;
    const int half = lane >> 4;
    const int n0   = wv * 16;

    const int ar = t >> 4;
    const int ac = (t & 15) * 2;
    const float* arow = g_ws + ((size_t)e * B_TOK + tile_m * 16 + ar) * FDIM;

    const size_t wbase = (size_t)e * FDIM * HDIM + h0;

    v8f acc = {};

    for (int k0 = 0; k0 < FDIM; k0 += 32) {
        float a0 = arow[k0 + ac];
        float a1 = arow[k0 + ac + 1];
        float4 bv[4];
#pragma unroll
        for (int i = 0; i < 4; ++i) {
            int q = t + i * 256;
            int row = q >> 5;
            int col = (q & 31) * 4;
            bv[i] = *(const float4*)(W2 + wbase + (size_t)(k0 + row) * HDIM + col);
        }
        __syncthreads();
        Alds[ar][ac] = a0; Alds[ar][ac + 1] = a1;
#pragma unroll
        for (int i = 0; i < 4; ++i) {
            int q = t + i * 256;
            int row = q >> 5;
            int col = (q & 31) * 4;
            *(float4*)&Blds[row][col] = bv[i];
        }
        __syncthreads();
#pragma unroll
        for (int kk = 0; kk < 8; ++kk) {
            const int kb = kk * 4 + half * 2;
            v2f a = *(const v2f*)&Alds[mi][kb];
            v2f b; b.x = Blds[kb][n0 + mi]; b.y = Blds[kb + 1][n0 + mi];
            acc = wmma4(a, b, acc);
        }
    }

    const size_t ybase = ((size_t)e * B_TOK + tile_m * 16) * HDIM + h0 + n0;
#pragma unroll
    for (int r = 0; r < 8; ++r) {
        const int m = r + half * 8;
        y_ws[ybase + (size_t)m * HDIM + mi] = acc[r];
    }
}

// ---------------------------------------------------------------------------
// Kernel 5: out[b][h] = w0 * y[e0][slot0][h] + w1 * y[e1][slot1][h]
// ---------------------------------------------------------------------------
__global__ void combine_kernel(const float* __restrict__ y_ws,
                               const int* __restrict__ tok_e,
                               const int* __restrict__ tok_slot,
                               const float* __restrict__ tok_w,
                               float* __restrict__ out) {
    const int idx = blockIdx.x * blockDim.x + threadIdx.x;  // 0 .. B*H-1
    const int b = idx >> 12;     // / 4096
    const int h = idx & 4095;
    const int e0 = tok_e[b * 2 + 0], e1 = tok_e[b * 2 + 1];
    const int s0 = tok_slot[b * 2 + 0], s1 = tok_slot[b * 2 + 1];
    const float w0 = tok_w[b * 2 + 0], w1 = tok_w[b * 2 + 1];
    const float v0 = y_ws[((size_t)e0 * B_TOK + s0) * HDIM + h];
    const float v1 = y_ws[((size_t)e1 * B_TOK + s1) * HDIM + h];
    out[idx] = w0 * v0 + w1 * v1;
}

// ---------------------------------------------------------------------------
// Host launcher.  Workspace layout (bytes):
//   [0,64)      int cnt[8]
//   [64,1088)   int list[8][32]
//   [1088,1344) int tok_e[32][2]
//   [1344,1600) int tok_slot[32][2]
//   [1600,1856) float tok_w[32][2]
//   [1856,2880) float logits[32][8]
//   [4096, +4MB)        float g_ws[8][32][4096]
//   [4096+4MB, +4MB)    float y_ws[8][32][4096]
// Total ~8.4 MB.
// ---------------------------------------------------------------------------
extern "C" void kernel_launch(void* const* d_in, const int* in_sizes, int n_in,
                              void* d_out, int out_size, void* d_ws, size_t ws_size,
                              hipStream_t stream) {
    const float* x  = (const float*)d_in[0];
    const float* Wg = (const float*)d_in[1];
    const float* W1 = (const float*)d_in[2];
    const float* W3 = (const float*)d_in[3];
    const float* W2 = (const float*)d_in[4];
    float* out = (float*)d_out;

    char* ws = (char*)d_ws;
    int*   cnt      = (int*)(ws + 0);
    int*   list     = (int*)(ws + 64);
    int*   tok_e    = (int*)(ws + 1088);
    int*   tok_slot = (int*)(ws + 1344);
    float* tok_w    = (float*)(ws + 1600);
    float* logits   = (float*)(ws + 1856);
    const size_t slab = (size_t)EDIM * B_TOK * FDIM * sizeof(float);  // 4 MB
    float* g_ws = (float*)(ws + 4096);
    float* y_ws = (float*)(ws + 4096 + slab);

    gate_logits_kernel<<<B_TOK, 256, 0, stream>>>(x, Wg, logits);
    gate_route_kernel<<<1, 64, 0, stream>>>(logits, cnt, list, tok_e, tok_slot, tok_w);

    dim3 g1(FDIM / 128, 2, EDIM);   // worst case: 32 tokens on one expert -> 2 M-tiles
    expert_gemm1_kernel<<<g1, 256, 0, stream>>>(x, W1, W3, cnt, list, g_ws);

    dim3 g2(HDIM / 128, 2, EDIM);
    expert_gemm2_kernel<<<g2, 256, 0, stream>>>(W2, cnt, g_ws, y_ws);

    combine_kernel<<<(B_TOK * HDIM) / 256, 256, 0, stream>>>(y_ws, tok_e, tok_slot, tok_w, out);
}